// GCNEncoder_63891933495450
// MI455X (gfx1250) — compile-verified
//
#include <hip/hip_runtime.h>

// ---------------------------------------------------------------------------
// 3-layer GCN for MI455X (gfx1250, wave32).
//   per layer:  h = relu?(x_in) @ W        (V_WMMA_F32_16X16X4_F32, K fully
//                                           unrolled, immediate-offset loads)
//               agg = h * dinv^2 + b       (self-loop + bias, float4 streams)
//               agg[dst] += h[src]*norm    (wave-per-edge, b128 loads + f32
//                                           global atomics, L2-resident)
// Degrees / dinv computed once and reused by all 3 layers.
// Working set (h, agg, edges ~58MB) fits in the 192MB L2; the scatter-add
// atomics are the roofline limiter, the 4.1 GFLOP of GEMM is negligible.
// ---------------------------------------------------------------------------

typedef __attribute__((ext_vector_type(2))) float v2f;
typedef __attribute__((ext_vector_type(4))) float v4f;
typedef __attribute__((ext_vector_type(8))) float v8f;

#define N_NODES 50000
#define N_EDGES 800000

__global__ void zero_f32(float* __restrict__ p, int n) {
  int i = blockIdx.x * blockDim.x + threadIdx.x;
  if (i < n) p[i] = 0.0f;
}

// deg[dst] += 1 per edge
__global__ void deg_scatter(const int* __restrict__ dst, float* __restrict__ deg, int E) {
  int e = blockIdx.x * blockDim.x + threadIdx.x;
  if (e < E) atomicAdd(&deg[dst[e]], 1.0f);
}

// dinv = rsqrt(deg + 1)   (+1 = self loop), in place
__global__ void finalize_dinv(float* __restrict__ d, int n) {
  int i = blockIdx.x * blockDim.x + threadIdx.x;
  if (i < n) d[i] = rsqrtf(d[i] + 1.0f);
}

// H[N,FOUT] = relu?(X[N,FIN]) @ W[FIN,FOUT], row-major, dims % 16 == 0.
// One wave computes one 16x16 output tile; compile-time FIN/FOUT -> the
// 32-step K-loop fully unrolls into immediate-offset global_load clauses
// feeding a v_wmma_f32_16x16x4_f32 accumulator chain.
template <bool RELU_IN, int FIN, int FOUT>
__global__ void gemm_wmma_f32(const float* __restrict__ X, const float* __restrict__ W,
                              float* __restrict__ H) {
  const int lane  = threadIdx.x;       // 0..31
  const int m     = lane & 15;         // row (A) / col (B) within tile
  const int hi    = lane >> 4;         // half-wave select (K pairing)
  const int tileM = blockIdx.x;
  const int tileN = blockIdx.y * blockDim.y + threadIdx.y;

  // ISA A layout (16x4 f32): lane L holds M=L%16; VGPR0 K=2*(L>=16), VGPR1 +1
  const float* __restrict__ arow = X + (size_t)(tileM * 16 + m) * FIN + 2 * hi;
  // B (4x16) mirrored layout: lane = N; rows K offset by 2*hi
  const float* __restrict__ wcol = W + (size_t)(2 * hi) * FOUT + tileN * 16 + m;

  v8f acc = {};
#pragma unroll
  for (int k0 = 0; k0 < FIN; k0 += 4) {
    v2f a = *(const v2f*)(arow + k0);                   // global_load_b64, imm offset
    if (RELU_IN) { a[0] = fmaxf(a[0], 0.0f); a[1] = fmaxf(a[1], 0.0f); }
    v2f b;
    b[0] = wcol[(size_t)k0 * FOUT];                     // imm-offset b32 loads
    b[1] = wcol[(size_t)(k0 + 1) * FOUT];
    acc = __builtin_amdgcn_wmma_f32_16x16x4_f32(
        /*neg_a=*/false, a, /*neg_b=*/false, b,
        /*c_mod=*/(short)0, acc, /*reuse_a=*/false, /*reuse_b=*/false);
  }

  // C/D layout: VGPR r -> row r (lanes 0-15) / row r+8 (lanes 16-31), col = lane%16
  float* __restrict__ hrow = H + (size_t)(tileM * 16 + hi * 8) * FOUT + tileN * 16 + m;
#pragma unroll
  for (int r = 0; r < 8; ++r) hrow[(size_t)r * FOUT] = acc[r];
}

// agg[i,f] = h[i,f] * dinv[i]^2 + b[f]  (self-loop + bias; float4 streams,
// fully overwrites the destination so no pre-zeroing is needed)
template <int F>
__global__ void init_agg(const float* __restrict__ h, const float* __restrict__ dinv,
                         const float* __restrict__ bias, float* __restrict__ agg, int nvec) {
  int i = blockIdx.x * blockDim.x + threadIdx.x;      // index in float4 units
  if (i >= nvec) return;
  constexpr int FV = F / 4;
  int node = i / FV;
  int fv = i - node * FV;
  float di = dinv[node];
  float d2 = di * di;
  v4f hv = *(const v4f*)(h + (size_t)i * 4);
  v4f bv = *(const v4f*)(bias + fv * 4);
  v4f o;
  o[0] = hv[0] * d2 + bv[0];
  o[1] = hv[1] * d2 + bv[1];
  o[2] = hv[2] * d2 + bv[2];
  o[3] = hv[3] * d2 + bv[3];
  *(v4f*)(agg + (size_t)i * 4) = o;
}

// One wave per edge: agg[dst,:] += h[src,:] * dinv[src]*dinv[dst]
// Each lane owns F/32 contiguous features: one b128/b64 load (512B coalesced
// per wave) feeding global_atomic_add_f32s into the L2-resident agg buffer.
template <int F>
__global__ void edge_scatter(const int* __restrict__ src, const int* __restrict__ dst,
                             const float* __restrict__ h, const float* __restrict__ dinv,
                             float* __restrict__ agg, int E) {
  int gid  = blockIdx.x * blockDim.x + threadIdx.x;
  int edge = gid >> 5;
  int lane = gid & 31;
  if (edge >= E) return;
  int s = src[edge];
  int d = dst[edge];
  float norm = dinv[s] * dinv[d];
  constexpr int PER_LANE = F / 32;                     // 4 (F=128) or 2 (F=64)
  const float* __restrict__ hs = h + (size_t)s * F + lane * PER_LANE;
  float* __restrict__ ad = agg + (size_t)d * F + lane * PER_LANE;
  if (PER_LANE == 4) {
    v4f hv = *(const v4f*)hs;                          // global_load_b128
#pragma unroll
    for (int j = 0; j < 4; ++j) atomicAdd(&ad[j], hv[j] * norm);
  } else {
    v2f hv = *(const v2f*)hs;                          // global_load_b64
#pragma unroll
    for (int j = 0; j < 2; ++j) atomicAdd(&ad[j], hv[j] * norm);
  }
}

extern "C" void kernel_launch(void* const* d_in, const int* in_sizes, int n_in,
                              void* d_out, int out_size, void* d_ws, size_t ws_size,
                              hipStream_t stream) {
  const float* x  = (const float*)d_in[0];
  const int* edges = (const int*)d_in[1];          // [2, E] flattened row-major
  const int* src = edges;
  const int* dst = edges + N_EDGES;
  const float* W1 = (const float*)d_in[2];
  const float* b1 = (const float*)d_in[3];
  const float* W2 = (const float*)d_in[4];
  const float* b2 = (const float*)d_in[5];
  const float* W3 = (const float*)d_in[6];
  const float* b3 = (const float*)d_in[7];
  float* out = (float*)d_out;

  // Workspace layout: dinv (200KB, padded) | bufA (25.6MB) | bufB (25.6MB)
  char* ws = (char*)d_ws;
  float* dinv = (float*)ws;
  float* bufA = (float*)(ws + 204800);
  float* bufB = (float*)(ws + 204800 + (size_t)N_NODES * 128 * sizeof(float));

  const int T = 256;
  // ---- degree -> dinv (shared by all layers) ----
  zero_f32<<<(N_NODES + T - 1) / T, T, 0, stream>>>(dinv, N_NODES);
  deg_scatter<<<(N_EDGES + T - 1) / T, T, 0, stream>>>(dst, dinv, N_EDGES);
  finalize_dinv<<<(N_NODES + T - 1) / T, T, 0, stream>>>(dinv, N_NODES);

  dim3 gblk(32, 4);                                  // 4 waves/block, 1 tile per wave
  int eblocks = (N_EDGES * 32 + T - 1) / T;          // wave-per-edge
  int v128 = N_NODES * 128 / 4;                      // float4 count, F=128
  int v64  = N_NODES * 64 / 4;                       // float4 count, F=64

  // ---- layer 1: h1 = x@W1 -> A ; agg1 -> B ----
  gemm_wmma_f32<false, 128, 128><<<dim3(N_NODES / 16, 2), gblk, 0, stream>>>(x, W1, bufA);
  init_agg<128><<<(v128 + T - 1) / T, T, 0, stream>>>(bufA, dinv, b1, bufB, v128);
  edge_scatter<128><<<eblocks, T, 0, stream>>>(src, dst, bufA, dinv, bufB, N_EDGES);

  // ---- layer 2: h2 = relu(agg1)@W2 -> A ; agg2 -> B (B dead after GEMM) ----
  gemm_wmma_f32<true, 128, 128><<<dim3(N_NODES / 16, 2), gblk, 0, stream>>>(bufB, W2, bufA);
  init_agg<128><<<(v128 + T - 1) / T, T, 0, stream>>>(bufA, dinv, b2, bufB, v128);
  edge_scatter<128><<<eblocks, T, 0, stream>>>(src, dst, bufA, dinv, bufB, N_EDGES);

  // ---- layer 3: h3 = relu(agg2)@W3 -> A ; agg3 -> d_out ----
  gemm_wmma_f32<true, 128, 64><<<dim3(N_NODES / 16, 1), gblk, 0, stream>>>(bufB, W3, bufA);
  init_agg<64><<<(v64 + T - 1) / T, T, 0, stream>>>(bufA, dinv, b3, out, v64);
  edge_scatter<64><<<eblocks, T, 0, stream>>>(src, dst, bufA, dinv, out, N_EDGES);
}